// Aggregator_4784593568023
// MI455X (gfx1250) — compile-verified
//
#include <hip/hip_runtime.h>

// GraphSAGE masked-mean aggregation on gfx1250 (MI455X).
// Formulated as D(16 feat x 16 node) = F_gathered(16 x K) * Select(K x 16),
// K = 256 (16 nodes x 16 samples), driven by V_WMMA_F32_16X16X4_F32.

typedef float v2f __attribute__((ext_vector_type(2)));
typedef float v8f __attribute__((ext_vector_type(8)));

#define D_FEAT    128
#define N_SAMPLES 16
#define NPB       16                 // nodes per block (one wave32 per block)
#define PAIRS     (NPB * N_SAMPLES)  // 256 (node,sample) pairs = K dimension

__global__ __launch_bounds__(32)
void sage_mean_wmma_f32(const float* __restrict__ feat,
                        const int*   __restrict__ nidx,
                        const int*   __restrict__ nmask,
                        float*       __restrict__ out,
                        int n_nodes) {
  __shared__ unsigned int s_idx[PAIRS];
  __shared__ float        s_w[PAIRS];
  __shared__ float        s_inv[NPB];

  const int lane  = threadIdx.x;          // 0..31
  const int nbase = blockIdx.x * NPB;

  // ---- Stage neighbor row ids + raw mask for 16 nodes into LDS ----
  for (int p = lane; p < PAIRS; p += 32) {
    const int node = nbase + (p >> 4);
    if (node < n_nodes) {
      const long long g = (long long)node * N_SAMPLES + (p & 15);
      s_idx[p] = (unsigned int)nidx[g];
      s_w[p]   = nmask[g] ? 1.0f : 0.0f;
    } else {
      s_idx[p] = 0u;
      s_w[p]   = 0.0f;
    }
  }
  __syncthreads();

  // ---- Per-node 1/max(cnt,1); nodes with cnt==0 keep all-zero weights ----
  if (lane < NPB) {
    float c = 0.0f;
    #pragma unroll
    for (int s = 0; s < N_SAMPLES; ++s) c += s_w[lane * N_SAMPLES + s];
    s_inv[lane] = 1.0f / fmaxf(c, 1.0f);
  }
  __syncthreads();

  for (int p = lane; p < PAIRS; p += 32) s_w[p] *= s_inv[p >> 4];
  __syncthreads();

  // ---- Accumulators: 8 feature tiles of D(16x16) f32 = 8 x v8f ----
  const v8f zero = {0.f, 0.f, 0.f, 0.f, 0.f, 0.f, 0.f, 0.f};
  v8f acc[8];
  #pragma unroll
  for (int t = 0; t < 8; ++t) acc[t] = zero;

  // A-matrix 16x4 f32 layout: lanes 0-15 hold K=0 (v0) / K=1 (v1),
  // lanes 16-31 hold K=2 / K=3. B 4x16 mirrored (lanes = columns).
  const int m    = lane & 15;             // feature-row (A) / node-column (B)
  const int koff = (lane >> 4) << 1;      // 0 for low half-wave, 2 for high

  #pragma unroll 2
  for (int ks = 0; ks < PAIRS / 4; ++ks) {   // 64 K-steps of 4 samples
    const int p0 = ks << 2;
    const int n  = ks >> 2;                  // the one node this K-step feeds

    const unsigned int r0 = s_idx[p0 + koff];
    const unsigned int r1 = s_idx[p0 + koff + 1];
    const float        w0 = s_w[p0 + koff];
    const float        w1 = s_w[p0 + koff + 1];

    v2f b;                                   // selection column: only c==n live
    b.x = (m == n) ? w0 : 0.0f;
    b.y = (m == n) ? w1 : 0.0f;

    const float* row0 = feat + (size_t)r0 * D_FEAT + m;
    const float* row1 = feat + (size_t)r1 * D_FEAT + m;

    #pragma unroll
    for (int t = 0; t < 8; ++t) {            // 8 feature tiles of 16
      v2f a;
      a.x = row0[t * 16];                    // 64B contiguous per half-wave
      a.y = row1[t * 16];
      acc[t] = __builtin_amdgcn_wmma_f32_16x16x4_f32(
          /*neg_a=*/false, a, /*neg_b=*/false, b,
          /*c_mod=*/(short)0, acc[t], /*reuse_a=*/false, /*reuse_b=*/false);
    }
  }

  // ---- Store: lane = node column; VGPR r = feature row r (+8 for hi half) ----
  const int node = nbase + m;
  const int fhi  = (lane >> 4) * 8;          // 0 or 8 within the 16-wide tile
  if (node < n_nodes) {
    float* dst_base = out + (size_t)node * D_FEAT + fhi;
    #pragma unroll
    for (int t = 0; t < 8; ++t) {
      *(v8f*)(dst_base + t * 16) = acc[t];   // 32B-aligned contiguous store
    }
  }
}

extern "C" void kernel_launch(void* const* d_in, const int* in_sizes, int n_in,
                              void* d_out, int out_size, void* d_ws, size_t ws_size,
                              hipStream_t stream) {
  const float* feat  = (const float*)d_in[0];
  const int*   nidx  = (const int*)d_in[1];
  const int*   nmask = (const int*)d_in[2];
  float*       out   = (float*)d_out;

  const int n_nodes = in_sizes[0] / D_FEAT;             // 100000
  const int blocks  = (n_nodes + NPB - 1) / NPB;        // 6250

  sage_mean_wmma_f32<<<blocks, 32, 0, stream>>>(feat, nidx, nmask, out, n_nodes);
}